// Cost_41953240547543
// MI455X (gfx1250) — compile-verified
//
#include <hip/hip_runtime.h>

// Stereo cost volume: out[B, 2C, D, H, W]
//   ch in [0,C):   left[b,c,h,w]   * (w >= d)
//   ch in [C,2C):  right[b,c,h,w-d]* (w >= d)
// Pure data movement (~403MB stores) -> optimize the CDNA5 data-movement path:
// async global->LDS staging (ASYNCcnt) + b128 non-temporal stores.

typedef float v4f __attribute__((ext_vector_type(4)));
typedef int   v4i __attribute__((ext_vector_type(4)));
typedef __attribute__((address_space(1))) v4i* gv4i_p;   // global int4*
typedef __attribute__((address_space(3))) v4i* lv4i_p;   // LDS int4*

#define NB 4
#define NC 32
#define NH 64
#define NW 128
#define ND 48
#define DTILE 8            // disparities per block
#define CH_STRIDE ((size_t)NH * NW)          // 8192 floats between channels
#define OUT_CH_STRIDE ((size_t)ND * NH * NW) // 393216
#define OUT_D_STRIDE  ((size_t)NH * NW)      // 8192

__device__ __forceinline__ void async_copy_b128(const float* gsrc, float* ldst) {
#if __has_builtin(__builtin_amdgcn_global_load_async_to_lds_b128)
  v4i* g_generic = (v4i*)gsrc;   // drop const + reinterpret in generic AS
  v4i* l_generic = (v4i*)ldst;
  __builtin_amdgcn_global_load_async_to_lds_b128(
      (gv4i_p)g_generic, (lv4i_p)l_generic,
      /*imm offset=*/0, /*cpol=*/0);
#else
  *(v4f*)ldst = *(const v4f*)gsrc;  // synchronous fallback
#endif
}

__device__ __forceinline__ void wait_async_zero() {
#if __has_builtin(__builtin_amdgcn_s_wait_asynccnt)
  __builtin_amdgcn_s_wait_asynccnt(0);
#else
  asm volatile("s_wait_asynccnt 0" ::: "memory");
#endif
}

__global__ __launch_bounds__(256) void cost_volume_kernel(
    const float* __restrict__ left, const float* __restrict__ right,
    float* __restrict__ out) {
  __shared__ float lleft[NC][NW];   // 16 KB
  __shared__ float lright[NC][NW];  // 16 KB

  const int tid = threadIdx.x;
  const int bh  = blockIdx.x;       // b*NH + h
  const int b   = bh >> 6;          // / NH
  const int h   = bh & (NH - 1);
  const int d0  = blockIdx.y * DTILE;

  // ---- Stage left/right rows for this (b, h) into LDS via async b128 loads.
  const float* gl = left  + (size_t)b * NC * CH_STRIDE + (size_t)h * NW;
  const float* gr = right + (size_t)b * NC * CH_STRIDE + (size_t)h * NW;

  for (int i = tid; i < NC * (NW / 4); i += 256) {   // 1024 float4 per buffer
    const int c  = i >> 5;            // i / 32
    const int w0 = (i & 31) << 2;     // (i % 32) * 4
    async_copy_b128(gl + (size_t)c * CH_STRIDE + w0, &lleft[c][w0]);
    async_copy_b128(gr + (size_t)c * CH_STRIDE + w0, &lright[c][w0]);
  }
  wait_async_zero();   // this wave's async writes to LDS are done
  __syncthreads();     // all waves' staging visible

  // ---- Emit DTILE disparity slices with non-temporal b128 stores.
  float* outb = out + (size_t)b * (2 * NC) * OUT_CH_STRIDE + (size_t)h * NW;

  for (int d = d0; d < d0 + DTILE; ++d) {
    for (int i = tid; i < 2 * NC * (NW / 4); i += 256) {  // 2048 float4 per d
      const int ch = i >> 5;          // 0..63
      const int w0 = (i & 31) << 2;   // 0,4,...,124
      const int c  = ch & (NC - 1);
      v4f v;
      if (ch < NC) {
        // left half: aligned vector load, mask the d-edge
        v = *(const v4f*)&lleft[c][w0];
        if (w0 < d) {
          v.x = (w0 + 0 >= d) ? v.x : 0.0f;
          v.y = (w0 + 1 >= d) ? v.y : 0.0f;
          v.z = (w0 + 2 >= d) ? v.z : 0.0f;
          v.w = (w0 + 3 >= d) ? v.w : 0.0f;
        }
      } else {
        // right half: shifted gather from LDS (OOB LDS reads return 0, and are
        // select-masked anyway)
        v.x = (w0 + 0 >= d) ? lright[c][w0 + 0 - d] : 0.0f;
        v.y = (w0 + 1 >= d) ? lright[c][w0 + 1 - d] : 0.0f;
        v.z = (w0 + 2 >= d) ? lright[c][w0 + 2 - d] : 0.0f;
        v.w = (w0 + 3 >= d) ? lright[c][w0 + 3 - d] : 0.0f;
      }
      float* dst = outb + (size_t)ch * OUT_CH_STRIDE + (size_t)d * OUT_D_STRIDE + w0;
      __builtin_nontemporal_store(v, (v4f*)dst);  // streaming store (write-once output)
    }
  }
}

extern "C" void kernel_launch(void* const* d_in, const int* in_sizes, int n_in,
                              void* d_out, int out_size, void* d_ws, size_t ws_size,
                              hipStream_t stream) {
  (void)in_sizes; (void)n_in; (void)out_size; (void)d_ws; (void)ws_size;
  const float* left  = (const float*)d_in[0];
  const float* right = (const float*)d_in[1];
  float* out = (float*)d_out;

  dim3 grid(NB * NH, ND / DTILE);   // (256, 6) blocks
  cost_volume_kernel<<<grid, 256, 0, stream>>>(left, right, out);
}